// DecoderRNN_8272107012785
// MI455X (gfx1250) — compile-verified
//
#include <hip/hip_runtime.h>
#include <hip/hip_bf16.h>
#include <math.h>

typedef __attribute__((ext_vector_type(2))) float v2f;
typedef __attribute__((ext_vector_type(8))) float v8f;

#define LDS_STRIDE 68   // 64 + 4: conflict-free strided tile reads, float4-aligned rows
#define V_VOCAB 50257
#define T_MAX 20

__device__ __forceinline__ v8f wmma_f32_k4(v2f a, v2f b, v8f c) {
  // D = A(16x4 f32) * B(4x16 f32) + C(16x16 f32), full fp32 precision
  return __builtin_amdgcn_wmma_f32_16x16x4_f32(false, a, false, b, (short)0, c, false, false);
}

// One wave computes D rows 0..63 (4 M-tiles) x cols [n0, n0+16), K = 512.
// A: 64x512 row-major global (staged in LDS by the whole block).
// B: 512 x ldb row-major global; column loads clamped to < ncolmax.
// Every wave in the block must call this (contains __syncthreads).
__device__ __forceinline__ void gemm_64xK512(
    const float* __restrict__ A, const float* __restrict__ B,
    int ldb, int n0, int ncolmax, float* sA, v8f acc[4]) {
  const int lane  = threadIdx.x & 31;
  const int lhalf = lane >> 4;
  const int lcol  = lane & 15;
  const int ncol  = min(n0 + lcol, ncolmax - 1);
  for (int kc = 0; kc < 512; kc += 64) {
    __syncthreads();
    for (int i = threadIdx.x; i < 64 * 16; i += blockDim.x) {
      int r  = i >> 4;
      int c4 = (i & 15) << 2;
      *(float4*)&sA[r * LDS_STRIDE + c4] = *(const float4*)&A[r * 512 + kc + c4];
    }
    __syncthreads();
#pragma unroll
    for (int kb = 0; kb < 16; ++kb) {
      int kl = (kb << 2) + (lhalf << 1);      // k within chunk for this half-wave
      int kg = kc + kl;
      const float* bp = B + (size_t)kg * ldb + ncol;
      v2f bfrag; bfrag.x = bp[0]; bfrag.y = bp[ldb];
#pragma unroll
      for (int mt = 0; mt < 4; ++mt) {
        v2f afrag = *(const v2f*)&sA[(mt * 16 + lcol) * LDS_STRIDE + kl];
        acc[mt] = wmma_f32_k4(afrag, bfrag, acc[mt]);
      }
    }
  }
}

// h0 = features @ W_hinit + b_hinit ; c0 = features @ W_cinit + b_cinit
// grid: 8 blocks x 256 threads (blocks 0-3 -> h0 tiles, 4-7 -> c0 tiles)
__global__ void k_init(const float* __restrict__ features,
                       const float* __restrict__ W_hinit, const float* __restrict__ b_hinit,
                       const float* __restrict__ W_cinit, const float* __restrict__ b_cinit,
                       float* __restrict__ h, float* __restrict__ c) {
  __shared__ __align__(16) float sA[64 * LDS_STRIDE];
  int wave = threadIdx.x >> 5;
  int lane = threadIdx.x & 31;
  int which = blockIdx.x >> 2;                 // 0 = h0, 1 = c0
  int tile  = (blockIdx.x & 3) * 8 + wave;     // 0..31
  int n0 = tile * 16;
  const float* W    = which ? W_cinit : W_hinit;
  const float* bias = which ? b_cinit : b_hinit;
  float* D          = which ? c : h;
  v8f acc[4] = {};
  gemm_64xK512(features, W, 512, n0, 512, sA, acc);
  int lhalf = lane >> 4, lcol = lane & 15;
  int col = n0 + lcol;
  float bv = bias[col];
#pragma unroll
  for (int mt = 0; mt < 4; ++mt)
#pragma unroll
    for (int i = 0; i < 8; ++i) {
      int m = mt * 16 + i + 8 * lhalf;
      D[m * 512 + col] = acc[mt][i] + bv;
    }
}

// Step part 1: beta-gate GEMM + x assembly.
// x[:, 512:] = sigmoid(h @ W_beta + b_beta) * features   (attention collapses: alpha == 1)
// x[:, :512] = emb_t  (features at t==0, else embed_table[captions[:, t-1]])
// grid: 4 blocks x 256 threads (32 waves = 32 column tiles)
__global__ void k_step1(const float* __restrict__ h, const float* __restrict__ features,
                        const float* __restrict__ W_beta, const float* __restrict__ b_beta,
                        const float* __restrict__ embed, const int* __restrict__ captions,
                        float* __restrict__ x, int t) {
  __shared__ __align__(16) float sA[64 * LDS_STRIDE];
  int wave = threadIdx.x >> 5;
  int lane = threadIdx.x & 31;
  int tile = blockIdx.x * 8 + wave;
  int n0 = tile * 16;
  v8f acc[4] = {};
  gemm_64xK512(h, W_beta, 512, n0, 512, sA, acc);
  int lhalf = lane >> 4, lcol = lane & 15;
  int col = n0 + lcol;
  float bv = b_beta[col];
#pragma unroll
  for (int mt = 0; mt < 4; ++mt)
#pragma unroll
    for (int i = 0; i < 8; ++i) {
      int m = mt * 16 + i + 8 * lhalf;
      float g = 1.0f / (1.0f + expf(-(acc[mt][i] + bv)));
      x[m * 1024 + 512 + col] = g * features[m * 512 + col];
    }
  // embedding half of x for this wave's 16 columns
  for (int m = lane; m < 64; m += 32) {
    const float* src = (t == 0) ? &features[m * 512]
                                : &embed[(size_t)captions[m * 19 + (t - 1)] * 512];
#pragma unroll
    for (int cc = 0; cc < 16; ++cc)
      x[m * 1024 + n0 + cc] = src[n0 + cc];
  }
}

// Step part 2: gates^T = [W_ih | W_hh] @ [x | h]^T  (M=2048 gate rows, N=64 batch, K=1536),
// fused with the LSTM cell update + length masking.
// grid: 32 blocks (one per 16 hidden dims) x 128 threads (wave w = gate w: i,f,g,o)
__global__ void k_step2(const float* __restrict__ x, float* __restrict__ h, float* __restrict__ c,
                        const float* __restrict__ W_ih, const float* __restrict__ W_hh,
                        const float* __restrict__ b_ih, const float* __restrict__ b_hh,
                        const int* __restrict__ lengths, int t) {
  __shared__ __align__(16) float sB[64 * LDS_STRIDE];  // [batch][k] chunk of [x|h]
  __shared__ float sg[4][16][64];                      // gate results [gate][hdim][batch]
  int lane  = threadIdx.x & 31;
  int wave  = threadIdx.x >> 5;        // gate index 0..3
  int lhalf = lane >> 4, lcol = lane & 15;
  int j  = blockIdx.x;                 // hidden-dim slice
  int g0 = j * 16 + wave * 512;        // base gate row for this wave
  v8f acc[4] = {};
  for (int kc = 0; kc < 1536; kc += 64) {
    __syncthreads();
    for (int i = threadIdx.x; i < 64 * 16; i += blockDim.x) {
      int r  = i >> 4;
      int c4 = (i & 15) << 2;
      int kg = kc + c4;
      const float* src = (kg < 1024) ? &x[r * 1024 + kg] : &h[r * 512 + (kg - 1024)];
      *(float4*)&sB[r * LDS_STRIDE + c4] = *(const float4*)src;
    }
    __syncthreads();
#pragma unroll
    for (int kb = 0; kb < 16; ++kb) {
      int kl = (kb << 2) + (lhalf << 1);
      int kg = kc + kl;
      v2f afrag;  // A = weight rows, per-lane row g0+lcol
      if (kg < 1024) {
        const float* p = &W_ih[(size_t)(g0 + lcol) * 1024 + kg];
        afrag.x = p[0]; afrag.y = p[1];
      } else {
        const float* p = &W_hh[(size_t)(g0 + lcol) * 512 + (kg - 1024)];
        afrag.x = p[0]; afrag.y = p[1];
      }
#pragma unroll
      for (int bt = 0; bt < 4; ++bt) {   // 4 batch tiles of 16
        v2f bfrag = *(const v2f*)&sB[(bt * 16 + lcol) * LDS_STRIDE + kl];
        acc[bt] = wmma_f32_k4(afrag, bfrag, acc[bt]);
      }
    }
  }
  __syncthreads();
#pragma unroll
  for (int bt = 0; bt < 4; ++bt)
#pragma unroll
    for (int i = 0; i < 8; ++i)
      sg[wave][i + 8 * lhalf][bt * 16 + lcol] = acc[bt][i];
  __syncthreads();
  // fused LSTM elementwise over 16 hidden dims x 64 batch
  for (int e = threadIdx.x; e < 16 * 64; e += blockDim.x) {
    int hd = e >> 6;
    int b  = e & 63;
    int gi = j * 16 + hd;
    float iv = sg[0][hd][b] + b_ih[gi]        + b_hh[gi];
    float fv = sg[1][hd][b] + b_ih[gi + 512]  + b_hh[gi + 512];
    float gv = sg[2][hd][b] + b_ih[gi + 1024] + b_hh[gi + 1024];
    float ov = sg[3][hd][b] + b_ih[gi + 1536] + b_hh[gi + 1536];
    float si = 1.0f / (1.0f + expf(-iv));
    float sf = 1.0f / (1.0f + expf(-fv));
    float so = 1.0f / (1.0f + expf(-ov));
    float cn = sf * c[b * 512 + gi] + si * tanhf(gv);
    float hn = so * tanhf(cn);
    if (t < lengths[b]) {          // frozen state past sequence end
      h[b * 512 + gi] = hn;
      c[b * 512 + gi] = cn;
    }
  }
}

// Step part 3 (dominant): preds = h @ W_fc + b_fc, masked to zero for t >= len.
// grid: ceil(3142/8)=393 blocks x 256 threads; one wave per 16-column vocab tile.
__global__ void k_step3(const float* __restrict__ h, const float* __restrict__ W_fc,
                        const float* __restrict__ b_fc, const int* __restrict__ lengths,
                        float* __restrict__ out, int t) {
  __shared__ __align__(16) float sA[64 * LDS_STRIDE];
  int wave = threadIdx.x >> 5;
  int lane = threadIdx.x & 31;
  const int ntiles = (V_VOCAB + 15) / 16;   // 3142
  int tile = blockIdx.x * 8 + wave;
  int tclamp = min(tile, ntiles - 1);
  int n0 = tclamp * 16;
  v8f acc[4] = {};
  gemm_64xK512(h, W_fc, V_VOCAB, n0, V_VOCAB, sA, acc);  // loads clamped to vocab
  if (tile >= ntiles) return;
  int lhalf = lane >> 4, lcol = lane & 15;
  int col = n0 + lcol;
  if (col >= V_VOCAB) return;
  float bv = b_fc[col];
#pragma unroll
  for (int mt = 0; mt < 4; ++mt)
#pragma unroll
    for (int i = 0; i < 8; ++i) {
      int m = mt * 16 + i + 8 * lhalf;
      float val = (t < lengths[m]) ? (acc[mt][i] + bv) : 0.0f;
      out[((size_t)m * T_MAX + t) * V_VOCAB + col] = val;
    }
}

extern "C" void kernel_launch(void* const* d_in, const int* in_sizes, int n_in,
                              void* d_out, int out_size, void* d_ws, size_t ws_size,
                              hipStream_t stream) {
  (void)in_sizes; (void)n_in; (void)out_size; (void)ws_size;
  const float* features = (const float*)d_in[0];
  const int*   captions = (const int*)d_in[1];
  const int*   lengths  = (const int*)d_in[2];
  const float* embed    = (const float*)d_in[3];
  const float* W_ih     = (const float*)d_in[4];
  const float* W_hh     = (const float*)d_in[5];
  const float* b_ih     = (const float*)d_in[6];
  const float* b_hh     = (const float*)d_in[7];
  // d_in[8..13] = W_enc/b_enc/W_dec/b_dec/W_full/b_full: dead (softmax over length-1 axis)
  const float* W_hinit  = (const float*)d_in[14];
  const float* b_hinit  = (const float*)d_in[15];
  const float* W_cinit  = (const float*)d_in[16];
  const float* b_cinit  = (const float*)d_in[17];
  const float* W_beta   = (const float*)d_in[18];
  const float* b_beta   = (const float*)d_in[19];
  const float* W_fc     = (const float*)d_in[20];
  const float* b_fc     = (const float*)d_in[21];
  float* out = (float*)d_out;

  float* h = (float*)d_ws;        // 64*512
  float* c = h + 64 * 512;        // 64*512
  float* x = c + 64 * 512;        // 64*1024  ([emb_t | gated awe])

  k_init<<<8, 256, 0, stream>>>(features, W_hinit, b_hinit, W_cinit, b_cinit, h, c);
  for (int t = 0; t < T_MAX; ++t) {
    k_step1<<<4, 256, 0, stream>>>(h, features, W_beta, b_beta, embed, captions, x, t);
    k_step2<<<32, 128, 0, stream>>>(x, h, c, W_ih, W_hh, b_ih, b_hh, lengths, t);
    k_step3<<<393, 256, 0, stream>>>(h, W_fc, b_fc, lengths, out, t);
  }
}